// LogicConv3d_74457553044330
// MI455X (gfx1250) — compile-verified
//
#include <hip/hip_runtime.h>
#include <hip/hip_bf16.h>
#include <stdint.h>

typedef __attribute__((ext_vector_type(2))) float        v2f;
typedef __attribute__((ext_vector_type(8))) float        v8f;
typedef __attribute__((ext_vector_type(4))) unsigned int v4u;
typedef __attribute__((ext_vector_type(8))) int          v8i;
typedef __attribute__((ext_vector_type(4))) int          v4i;

// Geometry (match reference)
//   x: (B=4, C=3, H=32, W=32, D=32) f32, output: (B, K=32, 30, 30, 30) f32
//   tile per block: c=3, h=3, w=10, d=32  -> 2880 floats in LDS
//   LUT rows: 31 nodes * 32 kernels = 992, 4 entries each.

__global__ __launch_bounds__(256) void logic_conv3d_kernel(
    const float* __restrict__ x, const int* __restrict__ kc,
    const float* __restrict__ w0, const float* __restrict__ w1,
    const float* __restrict__ w2, const float* __restrict__ w3,
    const float* __restrict__ w4, float* __restrict__ out)
{
    __shared__ __align__(16) float tile[2880];      // [c][h][w][d] = [3][3][10][32]
    __shared__ __align__(16) float luts[992 * 4];   // row = levelBase + node*32 + k
    __shared__ int offs[1024];                      // [rank][k][s] tile-local offsets

    const int tid = threadIdx.x;
    const int ow0 = blockIdx.x * 8;   // output-w tile base (0,8,16,24)
    const int oh  = blockIdx.y;       // 0..29
    const int b   = blockIdx.z;       // 0..3

    // ------------------------------------------------------------------
    // 1) Wave 0 issues one TDM tensor load: 4-D tile (d=32,w=10,h=3,c=3)
    //    from x[b, 0:3, oh:oh+3, ow0:ow0+10, 0:32] into LDS `tile`.
    //    Tensor dims = REMAINING extents from tile start so TDM
    //    zero-fills the (unused) out-of-range w lines.
    // ------------------------------------------------------------------
    if (tid < 32) {
        uint64_t ga = (uint64_t)(uintptr_t)
            (x + ((((uint64_t)b * 3u) * 32u + (unsigned)oh) * 32u + (unsigned)ow0) * 32u);
        v4u g0;
        g0.x = 1u;                                        // count=1, user descriptor
        g0.y = (unsigned)(uintptr_t)(&tile[0]);           // LDS byte offset (low 32 bits)
        g0.z = (unsigned)(ga & 0xffffffffull);            // global_addr[31:0]
        g0.w = (unsigned)((ga >> 32) & 0x1ffffffull)      // global_addr[56:32]
             | (2u << 30);                                // type = 2 ("image")
        uint64_t q0 = (2ull << 16)                        // data_size = 4 bytes
                    | (32ull << 48);                      // tensor_dim0 (d) = 32
        uint64_t q1 = ((uint64_t)(unsigned)(32 - ow0) << 16)  // tensor_dim1 (w remaining)
                    | (32ull << 48);                      // tile_dim0 (d) = 32
        uint64_t q2 = 10ull                               // tile_dim1 (w) = 10
                    | (3ull << 16)                        // tile_dim2 (h) = 3
                    | (32ull << 32);                      // tensor_dim0_stride = D = 32
        uint64_t q3 = (1024ull << 16);                    // tensor_dim1_stride = W*D = 1024
        v8i g1;
        g1[0] = (int)q0; g1[1] = (int)(q0 >> 32);
        g1[2] = (int)q1; g1[3] = (int)(q1 >> 32);
        g1[4] = (int)q2; g1[5] = (int)(q2 >> 32);
        g1[6] = (int)q3; g1[7] = (int)(q3 >> 32);
        v4i g2;
        g2.x = 32 - oh;        // tensor_dim2 (h remaining)
        g2.y = 3;              // tensor_dim3 (c) = 3
        g2.z = 32768;          // tensor_dim2_stride = H*W*D
        g2.w = (3 << 16);      // tile_dim3 (c) = 3
        v4i g3 = {0, 0, 0, 0};
        v8i g4 = {0, 0, 0, 0, 0, 0, 0, 0};
        __builtin_amdgcn_tensor_load_to_lds(g0, g1, g2, g3, g4, 0);
    }

    // ------------------------------------------------------------------
    // 2) Tile-local gather offsets (overlaps with the TDM transfer).
    //    kc layout: (2, K=32, S=16, 4) with comps (h, w, d, c).
    // ------------------------------------------------------------------
    for (int i = tid; i < 1024; i += 256) {
        int h = kc[4 * i + 0], w = kc[4 * i + 1];
        int d = kc[4 * i + 2], c = kc[4 * i + 3];
        offs[i] = ((c * 3 + h) * 10 + w) * 32 + d;
    }

    // ------------------------------------------------------------------
    // 3) LUT tables: softmax(w_row) @ GATES via WMMA f32 16x16x4.
    //    All values kept in NAMED scalars (no register arrays) so the
    //    A-fragment build is straight-line cndmask, not select trees.
    // ------------------------------------------------------------------
    {
        const int wave = tid >> 5;
        const int lane = tid & 31;
        const bool hi  = (lane >= 16);   // false: A cols 4c+0/1, true: 4c+2/3
        const int lrow = lane & 15;

        // B fragment is tile-invariant. GATES[g][n] = (g >> n) & 1.
        v2f Bfr[4];
        #pragma unroll
        for (int c4 = 0; c4 < 4; ++c4) {
            int g = 4 * c4 + (hi ? 2 : 0);
            Bfr[c4].x = (lrow < 4) ? (float)((g >> lrow) & 1) : 0.f;
            Bfr[c4].y = (lrow < 4) ? (float)(((g + 1) >> lrow) & 1) : 0.f;
        }

        for (int t = wave; t < 62; t += 8) {
            const int r0 = t * 16;
            const float* wp; int base;
            if (r0 < 512)      { wp = w0; base = 0;   }
            else if (r0 < 768) { wp = w1; base = 512; }
            else if (r0 < 896) { wp = w2; base = 768; }
            else if (r0 < 960) { wp = w3; base = 896; }
            else               { wp = w4; base = 960; }
            const float4* rp = (const float4*)(wp + (size_t)(r0 - base + lrow) * 16);
            const float4 ra = rp[0], rb = rp[1], rc = rp[2], rd = rp[3];

            float m = fmaxf(fmaxf(fmaxf(ra.x, ra.y), fmaxf(ra.z, ra.w)),
                     fmaxf(fmaxf(fmaxf(rb.x, rb.y), fmaxf(rb.z, rb.w)),
                     fmaxf(fmaxf(fmaxf(rc.x, rc.y), fmaxf(rc.z, rc.w)),
                           fmaxf(fmaxf(rd.x, rd.y), fmaxf(rd.z, rd.w)))));

            float e0  = __expf(ra.x - m), e1  = __expf(ra.y - m);
            float e2  = __expf(ra.z - m), e3  = __expf(ra.w - m);
            float e4  = __expf(rb.x - m), e5  = __expf(rb.y - m);
            float e6  = __expf(rb.z - m), e7  = __expf(rb.w - m);
            float e8  = __expf(rc.x - m), e9  = __expf(rc.y - m);
            float e10 = __expf(rc.z - m), e11 = __expf(rc.w - m);
            float e12 = __expf(rd.x - m), e13 = __expf(rd.y - m);
            float e14 = __expf(rd.z - m), e15 = __expf(rd.w - m);

            float sum = ((e0 + e1) + (e2 + e3)) + ((e4 + e5) + (e6 + e7))
                      + ((e8 + e9) + (e10 + e11)) + ((e12 + e13) + (e14 + e15));
            float inv = 1.f / sum;

            // A elements: lane half selects even/odd gate pair of each chunk.
            v2f A0, A1, A2, A3;
            A0.x = (hi ? e2  : e0 ) * inv;  A0.y = (hi ? e3  : e1 ) * inv;
            A1.x = (hi ? e6  : e4 ) * inv;  A1.y = (hi ? e7  : e5 ) * inv;
            A2.x = (hi ? e10 : e8 ) * inv;  A2.y = (hi ? e11 : e9 ) * inv;
            A3.x = (hi ? e14 : e12) * inv;  A3.y = (hi ? e15 : e13) * inv;

            v8f acc = {0.f, 0.f, 0.f, 0.f, 0.f, 0.f, 0.f, 0.f};
            acc = __builtin_amdgcn_wmma_f32_16x16x4_f32(false, A0, false, Bfr[0], (short)0, acc, false, false);
            acc = __builtin_amdgcn_wmma_f32_16x16x4_f32(false, A1, false, Bfr[1], (short)0, acc, false, false);
            acc = __builtin_amdgcn_wmma_f32_16x16x4_f32(false, A2, false, Bfr[2], (short)0, acc, false, false);
            acc = __builtin_amdgcn_wmma_f32_16x16x4_f32(false, A3, false, Bfr[3], (short)0, acc, false, false);

            if (lrow < 4) {                        // D cols 0..3 hold the 4 LUT entries
                #pragma unroll
                for (int v = 0; v < 8; ++v)
                    luts[(r0 + v + (hi ? 8 : 0)) * 4 + lrow] = acc[v];
            }
        }
    }

    if (tid < 32) __builtin_amdgcn_s_wait_tensorcnt(0);
    __syncthreads();

    // ------------------------------------------------------------------
    // 4) Gather from LDS tile + 5-level LUT tree, 32 kernels per thread.
    // ------------------------------------------------------------------
    const int owl = tid / 30;
    const int od  = tid % 30;
    const int ow  = ow0 + owl;
    if (tid < 240 && ow < 30) {
        const int toff = owl * 32 + od;

        auto lut_eval = [&](float a, float bb, int rowidx) -> float {
            const float4 l = *(const float4*)&luts[rowidx * 4];
            float na = 1.f - a, nb = 1.f - bb;
            return l.x * na * nb + l.y * na * bb + l.z * a * nb + l.w * a * bb;
        };

        #pragma unroll 1
        for (int k = 0; k < 32; ++k) {
            const int* o0 = &offs[k * 16];
            const int* o1 = &offs[512 + k * 16];
            float vbuf[16];
            #pragma unroll
            for (int s = 0; s < 16; ++s) {
                float a  = tile[o0[s] + toff];
                float bb = tile[o1[s] + toff];
                vbuf[s] = lut_eval(a, bb, s * 32 + k);            // level 0
            }
            #pragma unroll
            for (int n = 0; n < 8; ++n)
                vbuf[n] = lut_eval(vbuf[2*n], vbuf[2*n+1], 512 + n * 32 + k);
            #pragma unroll
            for (int n = 0; n < 4; ++n)
                vbuf[n] = lut_eval(vbuf[2*n], vbuf[2*n+1], 768 + n * 32 + k);
            #pragma unroll
            for (int n = 0; n < 2; ++n)
                vbuf[n] = lut_eval(vbuf[2*n], vbuf[2*n+1], 896 + n * 32 + k);
            vbuf[0] = lut_eval(vbuf[0], vbuf[1], 960 + k);

            out[((((size_t)b * 32 + k) * 30 + oh) * 30 + ow) * 30 + od] = vbuf[0];
        }
    }
}

extern "C" void kernel_launch(void* const* d_in, const int* in_sizes, int n_in,
                              void* d_out, int out_size, void* d_ws, size_t ws_size,
                              hipStream_t stream) {
    (void)in_sizes; (void)n_in; (void)d_ws; (void)ws_size; (void)out_size;
    const float* x  = (const float*)d_in[0];
    const int*   kc = (const int*)d_in[1];
    const float* w0 = (const float*)d_in[2];
    const float* w1 = (const float*)d_in[3];
    const float* w2 = (const float*)d_in[4];
    const float* w3 = (const float*)d_in[5];
    const float* w4 = (const float*)d_in[6];
    float* out = (float*)d_out;

    dim3 grid(4, 30, 4);   // (ow tiles of 8, oh, batch)
    logic_conv3d_kernel<<<grid, 256, 0, stream>>>(x, kc, w0, w1, w2, w3, w4, out);
}